// BackwardWarp_onnx_41798621725380
// MI455X (gfx1250) — compile-verified
//
#include <hip/hip_runtime.h>

// Backward bilinear warp, NCHW fp32: img (1,32,1024,2048), flow (1,2,1024,2048).
// Memory-bound gather/blend: ~0.55 GB moved, ~0.5 GFLOP -> pure HBM problem on
// MI455X (23.3 TB/s). No GEMM structure exists (gathered operand differs per
// pixel), so WMMA is intentionally not used; instead we optimize the memory
// path: b128 loads/stores, NT hints (keep img resident in 192MB L2),
// and gfx1250 global_prefetch_b8 across the 8MiB-strided channel planes.

typedef float v4f __attribute__((ext_vector_type(4)));  // native clang vector:
// accepted by __builtin_nontemporal_*; same 16B layout/alignment as float4.

static constexpr int Hc = 1024;
static constexpr int Wc = 2048;
static constexpr int Cc = 32;
static constexpr int HWc = Hc * Wc;
static constexpr int QUADS = HWc / 4;          // one thread handles 4 x-pixels
static constexpr int BLOCK = 256;              // 8 waves (wave32) per block

__global__ __launch_bounds__(BLOCK) void warp_bilinear_kernel(
    const float* __restrict__ img,
    const float* __restrict__ flow,
    float* __restrict__ out)
{
    const int q = blockIdx.x * BLOCK + threadIdx.x;   // 0 .. QUADS-1
    const int yy  = q >> 9;                           // W/4 = 512
    const int x4  = (q & 511) << 2;
    const int base = yy * Wc + x4;

    // Coalesced 128-bit flow loads; flow is read exactly once -> non-temporal.
    const v4f u4 = __builtin_nontemporal_load(
        reinterpret_cast<const v4f*>(flow + base));
    const v4f v4 = __builtin_nontemporal_load(
        reinterpret_cast<const v4f*>(flow + HWc + base));

    const float u[4] = {u4.x, u4.y, u4.z, u4.w};
    const float v[4] = {v4.x, v4.y, v4.z, v4.w};

    int   ia[4], ib[4], ic[4], id[4];
    float wa[4], wb[4], wcc[4], wd[4];

    #pragma unroll
    for (int j = 0; j < 4; ++j) {
        float x = (float)(x4 + j) + u[j];
        float y = (float)yy + v[j];
        // Replicate the reference's normalize/denormalize round trip exactly
        // (same op order -> same fp32 rounding).
        x = 2.0f * (x / (float)(Wc - 1) - 0.5f);
        y = 2.0f * (y / (float)(Hc - 1) - 0.5f);
        x = (x + 1.0f) * 0.5f * (float)(Wc - 1);
        y = (y + 1.0f) * 0.5f * (float)(Hc - 1);

        const float x0 = floorf(x);
        const float y0 = floorf(y);
        const float x1 = x0 + 1.0f;
        const float y1 = y0 + 1.0f;

        const int x0c = (int)fminf(fmaxf(x0, 0.0f), (float)(Wc - 1));
        const int x1c = (int)fminf(fmaxf(x1, 0.0f), (float)(Wc - 1));
        const int y0c = (int)fminf(fmaxf(y0, 0.0f), (float)(Hc - 1));
        const int y1c = (int)fminf(fmaxf(y1, 0.0f), (float)(Hc - 1));

        wa[j]  = (x1 - x) * (y1 - y);   // (x0c, y0c)
        wb[j]  = (x1 - x) * (y - y0);   // (x0c, y1c)
        wcc[j] = (x - x0) * (y1 - y);   // (x1c, y0c)
        wd[j]  = (x - x0) * (y - y0);   // (x1c, y1c)

        ia[j] = y0c * Wc + x0c;
        ib[j] = y1c * Wc + x0c;
        ic[j] = y0c * Wc + x1c;
        id[j] = y1c * Wc + x1c;
    }

    const float* __restrict__ p = img;
    float*       __restrict__ o = out + base;

    // 32 channel planes, 8 MiB apart. Unroll x2 -> up to 32 outstanding
    // gather loads per wave while keeping VGPR pressure / occupancy healthy.
    #pragma unroll 2
    for (int c = 0; c < Cc; ++c) {
        if (c + 4 < Cc) {
            // gfx1250 global_prefetch_b8: warm the gathered cachelines in the
            // plane 4 channels (32 MiB) ahead; covers the quad's span.
            __builtin_prefetch(p + 4 * HWc + ia[0], 0, 1);
            __builtin_prefetch(p + 4 * HWc + id[3], 0, 1);
        }

        v4f r;
        r.x = p[ia[0]] * wa[0] + p[ib[0]] * wb[0] + p[ic[0]] * wcc[0] + p[id[0]] * wd[0];
        r.y = p[ia[1]] * wa[1] + p[ib[1]] * wb[1] + p[ic[1]] * wcc[1] + p[id[1]] * wd[1];
        r.z = p[ia[2]] * wa[2] + p[ib[2]] * wb[2] + p[ic[2]] * wcc[2] + p[id[2]] * wd[2];
        r.w = p[ia[3]] * wa[3] + p[ib[3]] * wb[3] + p[ic[3]] * wcc[3] + p[id[3]] * wd[3];

        // Output is write-once streaming: non-temporal b128 store so the
        // 256 MiB result stream does not evict img from L2.
        __builtin_nontemporal_store(r, reinterpret_cast<v4f*>(o));

        p += HWc;
        o += HWc;
    }
}

extern "C" void kernel_launch(void* const* d_in, const int* in_sizes, int n_in,
                              void* d_out, int out_size, void* d_ws, size_t ws_size,
                              hipStream_t stream)
{
    (void)in_sizes; (void)n_in; (void)d_ws; (void)ws_size; (void)out_size;
    // setup_inputs order: img, flow, H, W (H/W fixed at 1024/2048, baked in).
    const float* img  = reinterpret_cast<const float*>(d_in[0]);
    const float* flow = reinterpret_cast<const float*>(d_in[1]);
    float*       out  = reinterpret_cast<float*>(d_out);

    dim3 grid(QUADS / BLOCK);   // 524288 / 256 = 2048 blocks
    dim3 block(BLOCK);
    warp_bilinear_kernel<<<grid, block, 0, stream>>>(img, flow, out);
}